// SetAbstractionLayer_39642548142389
// MI455X (gfx1250) — compile-verified
//
#include <hip/hip_runtime.h>
#include <hip/hip_bf16.h>

#define NBATCH   4
#define NPTS     32768
#define FEAT     131
#define NSAMP    1024
#define RAD2     0.1f        // reference compares d2 <= radius (0.1), not radius^2
#define NTILES   (NPTS / 16) // 2048 point tiles of 16

typedef __attribute__((ext_vector_type(2))) float v2f;
typedef __attribute__((ext_vector_type(8))) float v8f;

// ---------------------------------------------------------------------------
// Pack kernel: gather (x,y,z) out of the 131-float rows into a dense 3-float
// array in workspace. 1.5 MB total -> lives in L2 (192 MB) for both consumers.
// ---------------------------------------------------------------------------
__global__ void sa_pack_kernel(const float* __restrict__ x, float* __restrict__ pk) {
    size_t i = (size_t)blockIdx.x * blockDim.x + threadIdx.x;
    if (i >= (size_t)NBATCH * NPTS) return;
    const float* src = x + i * FEAT;
    float* dst = pk + i * 3;
    dst[0] = src[0]; dst[1] = src[1]; dst[2] = src[2];
}

// ---------------------------------------------------------------------------
// FPS: one workgroup (1024 threads = 32 waves) per batch. Sequential 1024
// steps; per step each thread updates 32 strided points' closest distance,
// then wave32 shuffle argmax + LDS cross-wave argmax (tie-break: lowest
// index, matching jnp.argmax). Latency-bound; STRIDE is compile-time so all
// addressing strength-reduces to incremental adds.
// ---------------------------------------------------------------------------
template <int STRIDE>
__global__ __launch_bounds__(1024) void sa_fps_kernel(const float* __restrict__ pts,
                                                      int* __restrict__ idx_ws,
                                                      float* __restrict__ idx_out) {
    const int b = blockIdx.x;
    const int tid = threadIdx.x;
    const int lane = tid & 31;
    const int wave = tid >> 5;
    const float* pb = pts + (size_t)b * NPTS * STRIDE;

    __shared__ float s_v[32];
    __shared__ int   s_i[32];
    __shared__ float s_cx, s_cy, s_cz;

    float closest[32];
#pragma unroll
    for (int k = 0; k < 32; ++k) closest[k] = __builtin_inff();

    if (tid == 0) { s_cx = pb[0]; s_cy = pb[1]; s_cz = pb[2]; }  // sel starts at 0
    __syncthreads();

    for (int s = 0; s < NSAMP; ++s) {
        const float cx = s_cx, cy = s_cy, cz = s_cz;
        float bv = -1.0f;
        int   bi = 0x7fffffff;
#pragma unroll
        for (int k = 0; k < 32; ++k) {
            const int p = tid + k * 1024;                 // ascending index per thread
            const float* pp = pb + (size_t)p * STRIDE;
            const float dx = pp[0] - cx, dy = pp[1] - cy, dz = pp[2] - cz;
            const float d2 = dx * dx + dy * dy + dz * dz;
            float c = closest[k];
            c = fminf(c, d2);
            closest[k] = c;
            if (c > bv) { bv = c; bi = p; }               // strict > keeps lowest index
        }
        // wave32 argmax reduction
#pragma unroll
        for (int off = 16; off > 0; off >>= 1) {
            const float ov = __shfl_down(bv, off);
            const int   oi = __shfl_down(bi, off);
            if (ov > bv || (ov == bv && oi < bi)) { bv = ov; bi = oi; }
        }
        if (lane == 0) { s_v[wave] = bv; s_i[wave] = bi; }
        __syncthreads();
        if (wave == 0) {
            float rv = s_v[lane];
            int   ri = s_i[lane];
#pragma unroll
            for (int off = 16; off > 0; off >>= 1) {
                const float ov = __shfl_down(rv, off);
                const int   oi = __shfl_down(ri, off);
                if (ov > rv || (ov == rv && oi < ri)) { rv = ov; ri = oi; }
            }
            if (lane == 0) {
                idx_ws[b * NSAMP + s]  = ri;
                idx_out[b * NSAMP + s] = (float)ri;
                const float* cp = pb + (size_t)ri * STRIDE;
                s_cx = cp[0]; s_cy = cp[1]; s_cz = cp[2];
            }
        }
        __syncthreads();
    }
}

// ---------------------------------------------------------------------------
// Ball-query counts via V_WMMA_F32_16X16X4_F32, with the full squared
// distance folded into the matrix op:
//   A (16x4)  : row M = (-2cx, -2cy, -2cz, 1)
//   B (4x16)  : col N = ( px,   py,   pz, |p|^2)
//   C (16x16) : element (M,N) = |c_M|^2   (bias, per-lane c2[] matches layout)
//   => D[M][N] = |c|^2 - 2 c.p + |p|^2 = d^2 directly.
// Epilogue per WMMA is just 8 cmp + 8 carry-adds. Fragment layouts per ISA
// 7.12.2 (A: K{0,1} lanes 0-15 / K{2,3} lanes 16-31; B/C/D: N = lane%16, row
// split across halves). One workgroup (8 waves) per (batch, 16-centroid
// tile); each wave strides the 2048 point tiles, unrolled x2 so two
// independent WMMAs cover the WMMA->VALU co-exec hazard. Uniform loop ->
// EXEC all-1s at every WMMA as required.
// ---------------------------------------------------------------------------
template <int STRIDE>
__global__ __launch_bounds__(256) void sa_ballq_kernel(const float* __restrict__ pts,
                                                       const int* __restrict__ idx,
                                                       float* __restrict__ counts_out) {
    const int b    = blockIdx.y;
    const int ct   = blockIdx.x;     // centroid tile, 0..63
    const int tid  = threadIdx.x;
    const int lane = tid & 31;
    const int wave = tid >> 5;       // 0..7
    const int col  = lane & 15;      // N / M position 0..15
    const bool hi  = lane >= 16;

    const float* pb = pts + (size_t)b * NPTS * STRIDE;

    __shared__ float s_csq[16];
    __shared__ int   s_part[8][16];

    // Centroid tile (same 16 centroids in every wave of this workgroup)
    const int ci = idx[b * NSAMP + ct * 16 + col];
    const float* cp = pb + (size_t)ci * STRIDE;
    const float cx = cp[0], cy = cp[1], cz = cp[2];
    if (tid < 16) s_csq[tid] = cx * cx + cy * cy + cz * cz;

    v2f afrag;
    afrag.x = hi ? (-2.0f * cz) : (-2.0f * cx);  // K=0 (lo) / K=2 (hi)
    afrag.y = hi ? 1.0f         : (-2.0f * cy);  // K=1 (lo) / K=3 (hi) = 1
    __syncthreads();

    v8f cbias;
#pragma unroll
    for (int r = 0; r < 8; ++r) cbias[r] = s_csq[r + (hi ? 8 : 0)];

    int acc[8] = {0, 0, 0, 0, 0, 0, 0, 0};

    for (int t = wave; t < NTILES; t += 16) {    // 128 iterations, x2 unrolled
#pragma unroll
        for (int u = 0; u < 2; ++u) {
            const int tt = t + u * 8;
            const int j  = tt * 16 + col;
            const float* pp = pb + (size_t)j * STRIDE;
            // prefetch this wave's tile two steps ahead (clamped -> valid addr)
            const int jn = (tt + 16 < NTILES) ? (j + 256) : j;
            __builtin_prefetch(pb + (size_t)jn * STRIDE, 0, 1);

            const float qx = pp[0], qy = pp[1], qz = pp[2];
            const float pn2 = qx * qx + qy * qy + qz * qz;

            v2f bfrag;
            bfrag.x = hi ? qz  : qx;   // K=0 (lo) / K=2 (hi)
            bfrag.y = hi ? pn2 : qy;   // K=1 (lo) / K=3 (hi) = |p|^2

            const v8f d = __builtin_amdgcn_wmma_f32_16x16x4_f32(
                /*neg_a=*/false, afrag, /*neg_b=*/false, bfrag,
                /*c_mod=*/(short)0, cbias, /*reuse_a=*/false, /*reuse_b=*/false);

#pragma unroll
            for (int r = 0; r < 8; ++r) acc[r] += (d[r] <= RAD2) ? 1 : 0;
        }
    }

    // Reduce within each 16-lane half (xor masks < 16 stay inside the half)
#pragma unroll
    for (int r = 0; r < 8; ++r) {
        int v = acc[r];
        v += __shfl_xor(v, 1);
        v += __shfl_xor(v, 2);
        v += __shfl_xor(v, 4);
        v += __shfl_xor(v, 8);
        acc[r] = v;
    }
    if (lane == 0) {
#pragma unroll
        for (int r = 0; r < 8; ++r) s_part[wave][r] = acc[r];      // M = 0..7
    }
    if (lane == 16) {
#pragma unroll
        for (int r = 0; r < 8; ++r) s_part[wave][r + 8] = acc[r];  // M = 8..15
    }
    __syncthreads();
    if (tid < 16) {
        int tot = 0;
#pragma unroll
        for (int w = 0; w < 8; ++w) tot += s_part[w][tid];
        counts_out[b * NSAMP + ct * 16 + tid] = (float)tot;
    }
}

// ---------------------------------------------------------------------------
extern "C" void kernel_launch(void* const* d_in, const int* in_sizes, int n_in,
                              void* d_out, int out_size, void* d_ws, size_t ws_size,
                              hipStream_t stream) {
    const float* x = (const float*)d_in[0];
    float* out = (float*)d_out;

    const size_t xElems = (size_t)NBATCH * NPTS * FEAT;
    float* out_idx = out + xElems;                 // 4*1024 floats (value-cast int32)
    float* out_cnt = out_idx + (size_t)NBATCH * NSAMP;

    // Workspace layout: [packed pts (4*32768*3 f32)] [idx (4*1024 i32)]
    // Fall back to reading strided coords straight out of x if ws is small.
    const size_t packElems = (size_t)NBATCH * NPTS * 3;
    const size_t idxBytes  = (size_t)NBATCH * NSAMP * sizeof(int);
    const bool usePacked = ws_size >= packElems * sizeof(float) + idxBytes;

    // Output 0: pass-through of x
    hipMemcpyAsync(d_out, x, xElems * sizeof(float), hipMemcpyDeviceToDevice, stream);

    dim3 grid(NSAMP / 16, NBATCH);
    if (usePacked) {
        float* ws_pack = (float*)d_ws;
        int* ws_idx = (int*)(ws_pack + packElems);
        const int tot = NBATCH * NPTS;
        sa_pack_kernel<<<(tot + 255) / 256, 256, 0, stream>>>(x, ws_pack);
        sa_fps_kernel<3><<<NBATCH, 1024, 0, stream>>>(ws_pack, ws_idx, out_idx);
        sa_ballq_kernel<3><<<grid, 256, 0, stream>>>(ws_pack, ws_idx, out_cnt);
    } else {
        int* ws_idx = (int*)d_ws;
        sa_fps_kernel<FEAT><<<NBATCH, 1024, 0, stream>>>(x, ws_idx, out_idx);
        sa_ballq_kernel<FEAT><<<grid, 256, 0, stream>>>(x, ws_idx, out_cnt);
    }
}